// BatchGroupItN_44676249813412
// MI455X (gfx1250) — compile-verified
//
#include <hip/hip_runtime.h>
#include <math.h>

// ---------------- problem constants ----------------
#define GRP   32
#define TNS   5
#define EPSV  1e-5f
#define NBLK  512              // (N*C)/32 / ... = 64*8 blocks of 32 channels
#define HW    3136             // 56*56
#define TILE_W 64
#define NSPLIT 7               // column splits per block
#define TILES_PER_WG 7         // 49 tiles per block = 7*7
#define GRID1 (NBLK * NSPLIT)  // 3584 workgroups
#define MCOLS (512.0f * 3136.0f)

typedef __attribute__((ext_vector_type(2))) float v2f;
typedef __attribute__((ext_vector_type(4))) float v4f;
typedef __attribute__((ext_vector_type(8))) float v8f;

// ---------------- workspace layout (floats) ----------------
// partials: GRID1 rows of 1056 = [0..1024) cross sums, [1024..1056) group sums
#define PART_STRIDE 1056
#define OFF_PART 0
#define OFF_RED  (GRID1 * PART_STRIDE)   // 3,784,704
#define OFF_WM   (OFF_RED + 1056)
#define OFF_BV   (OFF_WM + 1024)
// total ws requirement ~ (OFF_BV+32)*4 bytes ~= 15.2 MB

// =====================================================================
// Kernel 1: per-WG partial Gram matrix S[g1][g2] = sum x_g1 * x_g2 and
// per-group sums, via V_WMMA_F32_16X16X4_F32 over LDS-staged tiles.
// =====================================================================
__global__ __launch_bounds__(256)
void k1_cov(const float* __restrict__ x, float* __restrict__ part) {
    __shared__ float tile[32 * 68];   // pitch 68 -> conflict-free b64 operand reads
    __shared__ float redq[1024];      // quadrant half-combine
    __shared__ float srow[256];       // per-thread row sums

    const int t    = threadIdx.x;
    const int wg   = blockIdx.x;
    const int blk  = wg / NSPLIT;
    const int til0 = (wg % NSPLIT) * TILES_PER_WG;
    const float* xb = x + (size_t)blk * 32 * HW;

    // loader role: thread t -> row t/8, 8 consecutive cols
    const int lrow = t >> 3;
    const int lcol = (t & 7) * 8;

    // wave roles: 4 output quadrants x 2 column halves
    const int w    = t >> 5;
    const int lane = t & 31;
    const int q    = w & 3;
    const int qr   = (q >> 1) * 16;
    const int qc   = (q & 1) * 16;
    const int half = w >> 2;

    v8f   acc = {};
    float sum = 0.0f;

    for (int it = 0; it < TILES_PER_WG; ++it) {
        const int col0 = (til0 + it) * TILE_W;
        const float* src = xb + (size_t)lrow * HW + col0 + lcol;
        v4f a0 = *(const v4f*)(src);
        v4f a1 = *(const v4f*)(src + 4);
        *(v4f*)&tile[lrow * 68 + lcol]     = a0;
        *(v4f*)&tile[lrow * 68 + lcol + 4] = a1;
        sum += a0.x + a0.y + a0.z + a0.w + a1.x + a1.y + a1.z + a1.w;
        __syncthreads();

        // A 16x4: lane L holds rows M=L%16, K = (L/16)*2 + {0,1}
        // B 4x16: lane L holds cols N=L%16, K = (L/16)*2 + {0,1}
        const int kofs = half * 32 + ((lane >> 4) << 1);
        #pragma unroll
        for (int kk = 0; kk < 32; kk += 4) {
            v2f av = *(const v2f*)&tile[(qr + (lane & 15)) * 68 + kofs + kk];
            v2f bv = *(const v2f*)&tile[(qc + (lane & 15)) * 68 + kofs + kk];
            acc = __builtin_amdgcn_wmma_f32_16x16x4_f32(
                      false, av, false, bv, (short)0, acc, false, false);
        }
        __syncthreads();
    }

    // combine the two column halves of each quadrant through LDS
    if (half == 1) {
        #pragma unroll
        for (int r = 0; r < 8; ++r)
            redq[q * 256 + (((lane >> 4) * 8) + r) * 16 + (lane & 15)] = acc[r];
    }
    srow[t] = sum;
    __syncthreads();

    float* out = part + (size_t)wg * PART_STRIDE;
    if (half == 0) {
        #pragma unroll
        for (int r = 0; r < 8; ++r) {
            const int m = ((lane >> 4) * 8) + r;   // C/D: VGPR r -> M = r + 8*(lane/16)
            const int n = lane & 15;               //      N = lane%16
            out[(qr + m) * 32 + (qc + n)] = acc[r] + redq[q * 256 + m * 16 + n];
        }
    }
    if (t < 32) {
        float s = 0.0f;
        #pragma unroll
        for (int j = 0; j < 8; ++j) s += srow[t * 8 + j];
        out[1024 + t] = s;
    }
}

// =====================================================================
// Kernel 1r: deterministic reduction of the per-WG partials.
// =====================================================================
__global__ __launch_bounds__(256)
void k1r_reduce(const float* __restrict__ part, float* __restrict__ red) {
    const int t = blockIdx.x * blockDim.x + threadIdx.x;
    if (t >= PART_STRIDE) return;
    float s0 = 0.f, s1 = 0.f, s2 = 0.f, s3 = 0.f;
    for (int w = 0; w < GRID1; w += 4) {
        s0 += part[(size_t)(w + 0) * PART_STRIDE + t];
        s1 += part[(size_t)(w + 1) * PART_STRIDE + t];
        s2 += part[(size_t)(w + 2) * PART_STRIDE + t];
        s3 += part[(size_t)(w + 3) * PART_STRIDE + t];
    }
    red[t] = (s0 + s1) + (s2 + s3);
}

// =====================================================================
// Kernel 2: sigma, trace, Newton-Schulz (5 iters) on 32x32; emit
// WM = P*sqrt(1/tr) and folded mean vector b = WM @ mean.
// =====================================================================
__global__ __launch_bounds__(1024)
void k2_newton(const float* __restrict__ red,
               float* __restrict__ wmOut, float* __restrict__ bvOut) {
    __shared__ float mean[32];
    __shared__ float sigN[1024];
    __shared__ float P[1024];
    __shared__ float Ta[1024];
    __shared__ float Tb[1024];
    __shared__ float trinv_s;

    const int t = threadIdx.x;
    const int i = t >> 5, j = t & 31;
    const float minv = 1.0f / MCOLS;

    if (t < 32) mean[t] = red[1024 + t] * minv;
    __syncthreads();

    const float sij = red[t] * minv - mean[i] * mean[j] + (i == j ? EPSV : 0.0f);
    if (t == 0) {
        float tr = 0.0f;
        for (int g = 0; g < 32; ++g)
            tr += red[g * 33] * minv - mean[g] * mean[g] + EPSV;
        trinv_s = 1.0f / tr;
    }
    __syncthreads();
    const float trinv = trinv_s;
    sigN[t] = sij * trinv;
    P[t]    = (i == j) ? 1.0f : 0.0f;
    __syncthreads();

    for (int it = 0; it < TNS; ++it) {
        float a = 0.0f;
        #pragma unroll
        for (int k = 0; k < 32; ++k) a += P[i * 32 + k] * P[k * 32 + j];
        Ta[t] = a; __syncthreads();
        float b = 0.0f;
        #pragma unroll
        for (int k = 0; k < 32; ++k) b += Ta[i * 32 + k] * P[k * 32 + j];
        Tb[t] = b; __syncthreads();
        float c = 0.0f;
        #pragma unroll
        for (int k = 0; k < 32; ++k) c += Tb[i * 32 + k] * sigN[k * 32 + j];
        const float np = 1.5f * P[t] - 0.5f * c;
        __syncthreads();
        P[t] = np;
        __syncthreads();
    }

    const float wmv = P[t] * sqrtf(trinv);
    wmOut[t] = wmv;
    Ta[t] = wmv;
    __syncthreads();
    if (t < 32) {
        float b = 0.0f;
        #pragma unroll
        for (int g = 0; g < 32; ++g) b += Ta[t * 32 + g] * mean[g];
        bvOut[t] = b;
    }
}

// =====================================================================
// Kernel 3: y = WM @ x - b, then per-channel affine; WMMA with A (WM
// rows) preloaded in registers, B streamed from LDS tiles.
// =====================================================================
__global__ __launch_bounds__(256)
void k3_apply(const float* __restrict__ x, const float* __restrict__ wm,
              const float* __restrict__ bv, const float* __restrict__ weight,
              const float* __restrict__ bias, float* __restrict__ out) {
    __shared__ float tile[32 * 72];   // pitch 72 -> conflict-free B gathers
    __shared__ float lutW[32], lutB[32], lutV[32];

    const int t    = threadIdx.x;
    const int wg   = blockIdx.x;
    const int blk  = wg / NSPLIT;
    const int til0 = (wg % NSPLIT) * TILES_PER_WG;
    const int c8   = blk & 7;                       // blk = n*8 + c8 -> channel base c8*32
    const float* xb = x   + (size_t)blk * 32 * HW;
    float*       ob = out + (size_t)blk * 32 * HW;

    const int lane = t & 31;
    const int w    = t >> 5;
    const int qr   = (w & 1) * 16;   // output row quadrant
    const int qc   = (w >> 1) * 16;  // column sub-quadrant in 64-wide tile

    if (t < 32) {
        lutW[t] = weight[c8 * 32 + t];
        lutB[t] = bias[c8 * 32 + t];
        lutV[t] = bv[t];
    }

    // Preload A operands: A[m][k] = WM[qr+m][k]; lane L -> M=L%16, K=(L/16)*2+{0,1}
    v2f areg[8];
    #pragma unroll
    for (int s = 0; s < 8; ++s) {
        const int ko = s * 4 + ((lane >> 4) << 1);
        areg[s] = *(const v2f*)&wm[(qr + (lane & 15)) * 32 + ko];
    }
    __syncthreads();

    const int lrow = t >> 3;
    const int lcol = (t & 7) * 8;

    for (int it = 0; it < TILES_PER_WG; ++it) {
        const int col0 = (til0 + it) * TILE_W;
        const float* src = xb + (size_t)lrow * HW + col0 + lcol;
        v4f a0 = *(const v4f*)(src);
        v4f a1 = *(const v4f*)(src + 4);
        *(v4f*)&tile[lrow * 72 + lcol]     = a0;
        *(v4f*)&tile[lrow * 72 + lcol + 4] = a1;
        __syncthreads();

        v8f acc = {};
        #pragma unroll
        for (int s = 0; s < 8; ++s) {
            // B[k][n] = X[k][col]; lane L -> N=L%16, K=(L/16)*2+{0,1}
            const int krow = s * 4 + ((lane >> 4) << 1);
            v2f bvv;
            bvv.x = tile[(krow + 0) * 72 + qc + (lane & 15)];
            bvv.y = tile[(krow + 1) * 72 + qc + (lane & 15)];
            acc = __builtin_amdgcn_wmma_f32_16x16x4_f32(
                      false, areg[s], false, bvv, (short)0, acc, false, false);
        }

        #pragma unroll
        for (int r = 0; r < 8; ++r) {
            const int g = qr + ((lane >> 4) * 8) + r;   // D: M = r + 8*(lane/16)
            const int n = qc + (lane & 15);
            const float v = (acc[r] - lutV[g]) * lutW[g] + lutB[g];
            ob[(size_t)g * HW + col0 + n] = v;
        }
        __syncthreads();
    }
}

// =====================================================================
extern "C" void kernel_launch(void* const* d_in, const int* in_sizes, int n_in,
                              void* d_out, int out_size, void* d_ws, size_t ws_size,
                              hipStream_t stream) {
    const float* x      = (const float*)d_in[0];
    const float* weight = (const float*)d_in[1];
    const float* bias   = (const float*)d_in[2];
    float* out = (float*)d_out;
    float* ws  = (float*)d_ws;

    float* part = ws + OFF_PART;   // needs ~15.2 MB of d_ws
    float* red  = ws + OFF_RED;
    float* wmw  = ws + OFF_WM;
    float* bvw  = ws + OFF_BV;

    k1_cov<<<GRID1, 256, 0, stream>>>(x, part);
    k1r_reduce<<<(PART_STRIDE + 255) / 256, 256, 0, stream>>>(part, red);
    k2_newton<<<1, 1024, 0, stream>>>(red, wmw, bvw);
    k3_apply<<<GRID1, 256, 0, stream>>>(x, wmw, bvw, weight, bias, out);
    (void)in_sizes; (void)n_in; (void)out_size; (void)ws_size;
}